// MutagNet_61564061221146
// MI455X (gfx1250) — compile-verified
//
#include <hip/hip_runtime.h>
#include <hip/hip_bf16.h>

typedef __attribute__((ext_vector_type(2))) float v2f;
typedef __attribute__((ext_vector_type(8))) float v8f;

#define N_NODES  100000
#define N_EDGES  3200000
#define N_GRAPHS 5000
#define CH       32
#define HID      75
#define HIDP     80   // padded to 5 WMMA tiles

// ---------------------------------------------------------------- utilities
__global__ void fill_zero_kernel(float* __restrict__ p, int n) {
    int t = blockIdx.x * blockDim.x + threadIdx.x;
    if (t < n) p[t] = 0.0f;
}

// Build zero-padded W1p[2][32][80], W2p[2][80][32], b1p[2][80]
__global__ void prep_weights_kernel(const float* __restrict__ w1,
                                    const float* __restrict__ b1,
                                    const float* __restrict__ w2,
                                    float* __restrict__ W1p,
                                    float* __restrict__ W2p,
                                    float* __restrict__ b1p) {
    int t = blockIdx.x * blockDim.x + threadIdx.x;
    if (t < 2 * 32 * HIDP) {                       // W1p: [l][k][j]
        int j = t % HIDP, rem = t / HIDP;
        int k = rem % 32, l = rem / 32;
        W1p[t] = (j < HID) ? w1[l * 32 * HID + k * HID + j] : 0.0f;
    } else if (t < 2 * 32 * HIDP + 2 * HIDP * 32) { // W2p: [l][j][c]
        int u = t - 2 * 32 * HIDP;
        int c = u % 32, rem = u / 32;
        int j = rem % HIDP, l = rem / HIDP;
        W2p[u] = (j < HID) ? w2[l * HID * 32 + j * 32 + c] : 0.0f;
    } else if (t < 2 * 32 * HIDP + 2 * HIDP * 32 + 2 * HIDP) { // b1p: [l][j]
        int u = t - (2 * 32 * HIDP + 2 * HIDP * 32);
        int j = u % HIDP, l = u / HIDP;
        b1p[u] = (j < HID) ? b1[l * HID + j] : 0.0f;
    }
}

// ------------------------------------------------------------- node embed
// h[n][c] = x[n][:14] @ node_w[:,c] + node_b[c]     (wave covers one node)
__global__ void node_embed_kernel(const float* __restrict__ x,
                                  const float* __restrict__ w,
                                  const float* __restrict__ b,
                                  float* __restrict__ h) {
    int t = blockIdx.x * blockDim.x + threadIdx.x;
    if (t >= N_NODES * CH) return;
    int n = t >> 5, c = t & 31;
    float acc = b[c];
#pragma unroll
    for (int k = 0; k < 14; ++k) acc += x[n * 14 + k] * w[k * CH + c];
    h[t] = acc;
}

// ------------------------------------------------------------ edge message
// agg[dst] += relu(h[src] + edge_attr @ edge_w + edge_b)  (ea on the fly)
__global__ void edge_msg_kernel(const float* __restrict__ h,
                                const int*   __restrict__ ei,
                                const float* __restrict__ eattr,
                                const float* __restrict__ ew,
                                const float* __restrict__ eb,
                                float* __restrict__ agg) {
    const int c = threadIdx.x & 31;
    const float w0 = ew[c], w1 = ew[CH + c], w2 = ew[2 * CH + c], b = eb[c];
    const long long total = (long long)N_EDGES * CH;
    const long long stride = (long long)gridDim.x * blockDim.x;
    for (long long t = (long long)blockIdx.x * blockDim.x + threadIdx.x;
         t < total; t += stride) {
        int e = (int)(t >> 5);
        int src = ei[e];
        int dst = ei[N_EDGES + e];
        float a0 = eattr[e * 3 + 0];
        float a1 = eattr[e * 3 + 1];
        float a2 = eattr[e * 3 + 2];
        float m = h[src * CH + c] + (a0 * w0 + a1 * w1 + a2 * w2 + b);
        m = m > 0.0f ? m : 0.0f;
        atomicAdd(&agg[dst * CH + c], m);
    }
}

// ---------------------------------------------------------------- node MLP
// z2 = relu((h+agg) @ W1p + b1p) @ W2p + b2, plus per-channel sum/sumsq.
// One wave handles 16 nodes; fp32 WMMA 16x16x4.
__global__ void mlp_kernel(const float* __restrict__ h,
                           const float* __restrict__ agg,
                           float* __restrict__ z2,
                           const float* __restrict__ W1p,  // [32][80]
                           const float* __restrict__ W2p,  // [80][32]
                           const float* __restrict__ b1p,  // [80]
                           const float* __restrict__ b2,   // [32]
                           float* __restrict__ bnsum,
                           float* __restrict__ bnssq) {
    __shared__ float Tlds[8][16][84];   // per-wave 16x80 transpose staging
    const int lane = threadIdx.x & 31;
    const int wv   = threadIdx.x >> 5;
    const int base = (blockIdx.x * 8 + wv) * 16;
    if (base >= N_NODES) return;        // N_NODES % 16 == 0: whole waves only

    const int m    = lane & 15;         // A: M index / B,C,D: N index
    const int half = lane >> 4;
    const int koff = half * 2;          // A/B: lanes 16-31 hold K+2,K+3
    const int r    = base + m;

    // ---- A fragments for GEMM1 (K=32 -> 8 wmma steps), zin = h + agg
    v2f A[8];
#pragma unroll
    for (int kk = 0; kk < 8; ++kk) {
        int k0 = kk * 4 + koff;
        v2f ha = *(const v2f*)(h   + r * CH + k0);
        v2f ga = *(const v2f*)(agg + r * CH + k0);
        A[kk] = ha + ga;
    }

    // ---- GEMM1: 5 output tiles of 16 columns (80 padded), bias+relu -> LDS
#pragma unroll
    for (int n = 0; n < 5; ++n) {
        v8f acc = {};
        const int col = n * 16 + m;
#pragma unroll
        for (int kk = 0; kk < 8; ++kk) {
            int krow = kk * 4 + koff;
            v2f bfrag;
            bfrag.x = W1p[krow * HIDP + col];
            bfrag.y = W1p[(krow + 1) * HIDP + col];
            acc = __builtin_amdgcn_wmma_f32_16x16x4_f32(
                false, A[kk], false, bfrag, (short)0, acc, false, false);
        }
        const float bias = b1p[col];
#pragma unroll
        for (int i = 0; i < 8; ++i) {
            float v = acc[i] + bias;
            v = v > 0.0f ? v : 0.0f;
            Tlds[wv][i + half * 8][col] = v;   // C/D layout: M = i + 8*half
        }
    }

    // ---- GEMM2: [16x80] @ [80x32] -> 2 output tiles (wave-private LDS,
    //      same-wave DS ordering; compiler inserts s_wait_dscnt)
    v8f acc2[2] = {};
#pragma unroll
    for (int kk = 0; kk < 20; ++kk) {
        int k0 = kk * 4 + koff;
        v2f a = *(const v2f*)(&Tlds[wv][m][k0]);
#pragma unroll
        for (int n2 = 0; n2 < 2; ++n2) {
            v2f bfrag;
            bfrag.x = W2p[k0 * CH + n2 * 16 + m];
            bfrag.y = W2p[(k0 + 1) * CH + n2 * 16 + m];
            acc2[n2] = __builtin_amdgcn_wmma_f32_16x16x4_f32(
                false, a, false, bfrag, (short)0, acc2[n2], false, false);
        }
    }

    // ---- bias, store z2, accumulate BN statistics
#pragma unroll
    for (int n2 = 0; n2 < 2; ++n2) {
        const int ch = n2 * 16 + m;
        const float bias = b2[ch];
        float s = 0.0f, sq = 0.0f;
#pragma unroll
        for (int i = 0; i < 8; ++i) {
            float v = acc2[n2][i] + bias;
            z2[(base + i + half * 8) * CH + ch] = v;
            s += v; sq += v * v;
        }
        s  += __shfl_xor(s, 16);
        sq += __shfl_xor(sq, 16);
        if (half == 0) {
            atomicAdd(&bnsum[ch], s);
            atomicAdd(&bnssq[ch], sq);
        }
    }
}

// -------------------------------------------------------------- batch norm
__global__ void bn_finalize_kernel(float* __restrict__ bnsum,
                                   float* __restrict__ bnssq,
                                   const float* __restrict__ g,
                                   const float* __restrict__ b) {
    int c = threadIdx.x;
    if (c >= CH) return;
    const float inv = 1.0f / (float)N_NODES;
    float mu  = bnsum[c] * inv;
    float var = bnssq[c] * inv - mu * mu;
    float rs  = rsqrtf(var + 1e-5f);
    float sc  = rs * g[c];
    bnsum[c] = sc;               // scale
    bnssq[c] = b[c] - mu * sc;   // shift
}

__global__ void bn_apply_kernel(const float* __restrict__ z2,
                                const float* __restrict__ scale,
                                const float* __restrict__ shift,
                                float* __restrict__ h) {
    int t = blockIdx.x * blockDim.x + threadIdx.x;
    if (t >= N_NODES * CH) return;
    int c = t & 31;
    float v = z2[t] * scale[c] + shift[c];
    h[t] = v > 0.0f ? v : 0.0f;
}

// ---------------------------------------------------------------- pooling
__global__ void pool_kernel(const float* __restrict__ h,
                            const int* __restrict__ batch,
                            float* __restrict__ gsum,
                            float* __restrict__ gcnt) {
    int t = blockIdx.x * blockDim.x + threadIdx.x;
    if (t >= N_NODES * CH) return;
    int n = t >> 5, c = t & 31;
    int g = batch[n];
    atomicAdd(&gsum[g * CH + c], h[t]);
    if (c == 0) atomicAdd(&gcnt[g], 1.0f);
}

// -------------------------------------------------------------------- head
__global__ void head_kernel(const float* __restrict__ gsum,
                            const float* __restrict__ gcnt,
                            const float* __restrict__ w1,
                            const float* __restrict__ b1,
                            const float* __restrict__ w2,
                            const float* __restrict__ b2,
                            float* __restrict__ out) {
    int g = blockIdx.x * blockDim.x + threadIdx.x;
    if (g >= N_GRAPHS) return;
    float cnt = gcnt[g];
    cnt = cnt > 1.0f ? cnt : 1.0f;
    const float inv = 1.0f / cnt;
    float gx[CH];
#pragma unroll
    for (int c = 0; c < CH; ++c) gx[c] = gsum[g * CH + c] * inv;
    float o0 = b2[0], o1 = b2[1];
#pragma unroll
    for (int j = 0; j < 16; ++j) {
        float t = b1[j];
#pragma unroll
        for (int c = 0; c < CH; ++c) t += gx[c] * w1[c * 16 + j];
        t = t > 0.0f ? t : 0.0f;
        o0 += t * w2[j * 2 + 0];
        o1 += t * w2[j * 2 + 1];
    }
    out[g * 2 + 0] = o0;
    out[g * 2 + 1] = o1;
}

// ------------------------------------------------------------------ launch
extern "C" void kernel_launch(void* const* d_in, const int* in_sizes, int n_in,
                              void* d_out, int out_size, void* d_ws, size_t ws_size,
                              hipStream_t stream) {
    const float* x      = (const float*)d_in[0];
    const int*   eidx   = (const int*)  d_in[1];
    const float* eattr  = (const float*)d_in[2];
    const int*   batch  = (const int*)  d_in[3];
    const float* node_w = (const float*)d_in[4];
    const float* node_b = (const float*)d_in[5];
    const float* edge_w = (const float*)d_in[6];
    const float* edge_b = (const float*)d_in[7];
    const float* cw1    = (const float*)d_in[8];
    const float* cb1    = (const float*)d_in[9];
    const float* cw2    = (const float*)d_in[10];
    const float* cb2    = (const float*)d_in[11];
    const float* bn_g   = (const float*)d_in[12];
    const float* bn_b   = (const float*)d_in[13];
    const float* l1w    = (const float*)d_in[14];
    const float* l1b    = (const float*)d_in[15];
    const float* l2w    = (const float*)d_in[16];
    const float* l2b    = (const float*)d_in[17];
    float* out = (float*)d_out;

    // workspace layout (floats)
    float* ws    = (float*)d_ws;
    float* h     = ws;                           // N*32
    float* agg   = h   + (size_t)N_NODES * CH;   // N*32
    float* z2    = agg + (size_t)N_NODES * CH;   // N*32
    float* W1p   = z2  + (size_t)N_NODES * CH;   // 2*32*80
    float* W2p   = W1p + 2 * 32 * HIDP;          // 2*80*32
    float* b1p   = W2p + 2 * HIDP * 32;          // 2*80
    float* bnsum = b1p + 2 * HIDP;               // 32
    float* bnssq = bnsum + CH;                   // 32
    float* gsum  = bnssq + CH;                   // G*32
    float* gcnt  = gsum + (size_t)N_GRAPHS * CH; // G

    const int NC       = N_NODES * CH;           // 3.2M
    const int nc_grid  = (NC + 255) / 256;       // 12500

    prep_weights_kernel<<<(2 * 32 * HIDP + 2 * HIDP * 32 + 2 * HIDP + 255) / 256,
                          256, 0, stream>>>(cw1, cb1, cw2, W1p, W2p, b1p);
    node_embed_kernel<<<nc_grid, 256, 0, stream>>>(x, node_w, node_b, h);

    for (int l = 0; l < 2; ++l) {
        fill_zero_kernel<<<nc_grid, 256, 0, stream>>>(agg, NC);
        fill_zero_kernel<<<1, 64, 0, stream>>>(bnsum, 64);   // bnsum+bnssq
        edge_msg_kernel<<<8192, 256, 0, stream>>>(h, eidx, eattr,
                                                  edge_w, edge_b, agg);
        mlp_kernel<<<(N_NODES / 16 + 7) / 8, 256, 0, stream>>>(
            h, agg, z2,
            W1p + l * 32 * HIDP, W2p + l * HIDP * 32, b1p + l * HIDP,
            cb2 + l * CH, bnsum, bnssq);
        bn_finalize_kernel<<<1, 32, 0, stream>>>(bnsum, bnssq,
                                                 bn_g + l * CH, bn_b + l * CH);
        bn_apply_kernel<<<nc_grid, 256, 0, stream>>>(z2, bnsum, bnssq, h);
    }

    fill_zero_kernel<<<(N_GRAPHS * CH + N_GRAPHS + 255) / 256, 256, 0, stream>>>(
        gsum, N_GRAPHS * CH + N_GRAPHS);   // gsum+gcnt contiguous
    pool_kernel<<<nc_grid, 256, 0, stream>>>(h, batch, gsum, gcnt);
    head_kernel<<<(N_GRAPHS + 255) / 256, 256, 0, stream>>>(
        gsum, gcnt, l1w, l1b, l2w, l2b, out);
}